// DFFN_AutoCorr_17042430230757
// MI455X (gfx1250) — compile-verified
//
#include <hip/hip_runtime.h>
#include <math.h>
#include <stdint.h>

// ---------------------------------------------------------------------------
// DFFN + patchwise spectral autocorrelation, fused for MI455X (gfx1250).
//
// Math: irfft2(Xf + a*P) + b*irfft2(P) == irfft2(Xf + (a+b)*P)  (linearity),
// so the spectral block is ONE forward 8x8 rfft2, a per-bin op, ONE inverse.
// DFTs are unrolled matrix transforms with constexpr twiddles in {0,±1,±√½}
// so the compiler constant-folds ~half of the MACs.
//
// GEMMs (the two 1x1 convs) use V_WMMA_F32_16X16X4_F32 (native fp32 WMMA).
// Global->LDS staging uses GLOBAL_LOAD_ASYNC_TO_LDS_B32 (ASYNCcnt-tracked,
// no VGPR round-trip), with explicit s_wait_asynccnt before barriers.
// ---------------------------------------------------------------------------

typedef __attribute__((ext_vector_type(2))) float v2f;
typedef __attribute__((ext_vector_type(8))) float v8f;

#define HH   256
#define WW   256
#define CIN  64
#define CMID 256   // 2*hidden
#define CHID 128
#define COUT 64

// Kernel-1 LDS partition (floats): regionA = max(xs 64*64, fw 256*41); y1s padded 65
#define K1_REGA  10496
#define K1_Y1S   (256*65)
#define K1_LDS_F (K1_REGA + K1_Y1S)
// Kernel-2 LDS partition (floats)
#define K2_TS    (100*257)
#define K2_WD    2304
#define K2_GS    (128*65)
#define K2_LDS_F (K2_TS + K2_WD + K2_GS)

__device__ __forceinline__ v8f wmma4(v2f a, v2f b, v8f c) {
  // 8 args: (neg_a, A, neg_b, B, c_mod, C, reuse_a, reuse_b)
  return __builtin_amdgcn_wmma_f32_16x16x4_f32(false, a, false, b, (short)0, c,
                                               false, false);
}

// Async direct copy global -> LDS (one b32 per lane). ldsptr/gptr are per-lane.
// LDS aperture maps addr[31:0] to the LDS byte offset, so truncation is the
// wave-relative dsaddr the instruction expects.
__device__ __forceinline__ void async_g2lds_b32(const float* gptr, float* ldsptr) {
  unsigned ldsoff = (unsigned)(uintptr_t)ldsptr;
  unsigned long long ga = (unsigned long long)(uintptr_t)gptr;
  asm volatile("global_load_async_to_lds_b32 %0, %1, off"
               :: "v"(ldsoff), "v"(ga)
               : "memory");
}
__device__ __forceinline__ void wait_async0() {
  asm volatile("s_wait_asynccnt 0x0" ::: "memory");
}

// ============================ Kernel 1 =====================================
// project_in (WMMA GEMM 256x64x64 per patch) + spectral block.
// One workgroup (256 thr / 8 waves) per (batch, 8x8 patch).
__global__ __launch_bounds__(256) void k1_projin_spectral(
    const float* __restrict__ x, const float* __restrict__ w_in,
    const float* __restrict__ fft_w, const float* __restrict__ alpha_p,
    const float* __restrict__ beta_p, float* __restrict__ y2) {
  extern __shared__ float lds[];
  float* regionA = lds;            // phase 1: xs[64ch][64pix]; phase 2: fw[256][41]
  float* y1s = lds + K1_REGA;      // [256ch][65] padded -> bank=(ch+pix)%64

  const int t = threadIdx.x;
  const int blk = blockIdx.x;
  const int b = blk >> 10;
  const int prow = (blk >> 5) & 31;
  const int pcol = blk & 31;
  const int r0 = prow * 8, c0 = pcol * 8;

  // ---- stage input patch via async copy: xs[c][pix], pix = 8*r + cc ----
  {
    const float* xb = x + (size_t)b * CIN * HH * WW;
#pragma unroll
    for (int i = 0; i < 16; i++) {
      int lin = i * 256 + t;  // 0..4095
      int c = lin >> 6, pix = lin & 63;
      const float* gp =
          xb + (size_t)c * (HH * WW) + (size_t)(r0 + (pix >> 3)) * WW + (c0 + (pix & 7));
      async_g2lds_b32(gp, regionA + lin);
    }
  }
  wait_async0();
  __syncthreads();

  // ---- GEMM: y1[256][64] = w_in[256][64] * xs[64][64] via WMMA f32 16x16x4
  const int wave = t >> 5, lane = t & 31;
  const int lhi = lane >> 4, llo = lane & 15;  // lane-half / lane-low
  v8f acc[2][4];
#pragma unroll
  for (int mi = 0; mi < 2; mi++)
#pragma unroll
    for (int nt = 0; nt < 4; nt++)
#pragma unroll
      for (int r = 0; r < 8; r++) acc[mi][nt][r] = 0.f;

  for (int kk = 0; kk < 16; kk++) {
    // B frags (shared across M-tiles). Assumed B layout: lane=N, K = 2*j + lhi.
    v2f bf[4];
#pragma unroll
    for (int nt = 0; nt < 4; nt++) {
      int N0 = nt * 16 + llo;
      bf[nt].x = regionA[(kk * 4 + lhi) * 64 + N0];
      bf[nt].y = regionA[(kk * 4 + 2 + lhi) * 64 + N0];
    }
#pragma unroll
    for (int mi = 0; mi < 2; mi++) {
      // A layout (ISA table): lane = M (mod 16), K = j + 2*lhi  -> float2 load
      int M0 = (wave * 2 + mi) * 16 + llo;
      const float* wp = w_in + M0 * 64 + kk * 4 + 2 * lhi;
      v2f a;
      a.x = wp[0];
      a.y = wp[1];
#pragma unroll
      for (int nt = 0; nt < 4; nt++) acc[mi][nt] = wmma4(a, bf[nt], acc[mi][nt]);
    }
  }
  // D layout (ISA table): vgpr r, lanes0-15: M=r, lanes16-31: M=r+8; N=llo
#pragma unroll
  for (int mi = 0; mi < 2; mi++)
#pragma unroll
    for (int nt = 0; nt < 4; nt++)
#pragma unroll
      for (int r = 0; r < 8; r++) {
        int row = (wave * 2 + mi) * 16 + r + 8 * lhi;
        int col = nt * 16 + llo;
        y1s[row * 65 + col] = acc[mi][nt][r];
      }
  __syncthreads();  // GEMM done reading xs; y1s complete

  // ---- stage fft_w [256][40] -> fw stride 41 (conflict-free) ----
  for (int i = t; i < 256 * 40; i += 256)
    async_g2lds_b32(fft_w + i, regionA + (i / 40) * 41 + (i % 40));
  wait_async0();
  __syncthreads();

  // ---- spectral block, thread t = channel t ----
  const float sAB = alpha_p[0] + beta_p[0];
  constexpr float RS = 0.70710678118654752440f;
  const float C8[8] = {1.f, RS, 0.f, -RS, -1.f, -RS, 0.f, RS};
  const float S8[8] = {0.f, RS, 1.f, RS, 0.f, -RS, -1.f, -RS};
  const float* rowp = y1s + t * 65;
  const float* fwp = regionA + t * 41;

  float Fr[8][5], Fi[8][5];
  // forward along n (rfft rows): F[m][v] = sum_n p[m][n] e^{-2pi i v n/8}
#pragma unroll
  for (int m = 0; m < 8; m++) {
    float pr[8];
#pragma unroll
    for (int n = 0; n < 8; n++) pr[n] = rowp[m * 8 + n];
#pragma unroll
    for (int v = 0; v < 5; v++) {
      float sr = 0.f, si = 0.f;
#pragma unroll
      for (int n = 0; n < 8; n++) {
        const int a = (v * n) & 7;
        sr += pr[n] * C8[a];
        si -= pr[n] * S8[a];
      }
      Fr[m][v] = sr;
      Fi[m][v] = si;
    }
  }
  // per column v: fwd DFT over m, bin op (weight + power), inverse DFT over u
#pragma unroll
  for (int v = 0; v < 5; v++) {
    float Xr[8], Xi[8];
#pragma unroll
    for (int u = 0; u < 8; u++) {
      float sr = 0.f, si = 0.f;
#pragma unroll
      for (int m = 0; m < 8; m++) {
        const int a = (u * m) & 7;
        sr += Fr[m][v] * C8[a] + Fi[m][v] * S8[a];
        si += Fi[m][v] * C8[a] - Fr[m][v] * S8[a];
      }
      const float w = fwp[u * 5 + v];        // learned spectral weight
      const float xr = sr * w, xi = si * w;  // Xf = w * rfft2(p)
      Xr[u] = xr + sAB * (xr * xr + xi * xi);  // Y = Xf + (a+b)*|Xf|^2
      Xi[u] = xi;
    }
#pragma unroll
    for (int m = 0; m < 8; m++) {  // z[m][v] = (1/8) sum_u Y[u][v] e^{+2pi i u m/8}
      float sr = 0.f, si = 0.f;
#pragma unroll
      for (int u = 0; u < 8; u++) {
        const int a = (u * m) & 7;
        sr += Xr[u] * C8[a] - Xi[u] * S8[a];
        si += Xi[u] * C8[a] + Xr[u] * S8[a];
      }
      Fr[m][v] = sr * 0.125f;  // overwrite in place with z
      Fi[m][v] = si * 0.125f;
    }
  }
  // inverse rfft along n + store channel-last (coalesced across 256 channels)
#pragma unroll
  for (int m = 0; m < 8; m++) {
    float* outp = y2 + ((size_t)(b * HH + (r0 + m)) * WW + c0) * CMID + t;
#pragma unroll
    for (int n = 0; n < 8; n++) {
      float acc2 = Fr[m][0] + ((n & 1) ? -Fr[m][4] : Fr[m][4]);
#pragma unroll
      for (int v = 1; v < 4; v++) {
        const int a = (v * n) & 7;
        acc2 += 2.f * (Fr[m][v] * C8[a] - Fi[m][v] * S8[a]);
      }
      outp[(size_t)n * CMID] = 0.125f * acc2;
    }
  }
}

// ============================ Kernel 2 =====================================
// depthwise 3x3 (SAME, zero pad) + GLU(exact gelu) + project_out (WMMA GEMM).
__global__ __launch_bounds__(256) void k2_dw_glu_projout(
    const float* __restrict__ y2, const float* __restrict__ w_dw,
    const float* __restrict__ w_out, float* __restrict__ out) {
  extern __shared__ float lds[];
  float* ts = lds;                 // [100 halo pix][257 pad] bank=(pix+ch)%64
  float* wd = lds + K2_TS;         // [256][9]
  float* gs = lds + K2_TS + K2_WD; // g: [128][65 pad]

  const int t = threadIdx.x;
  const int blk = blockIdx.x;
  const int b = blk >> 10;
  const int prow = (blk >> 5) & 31;
  const int pcol = blk & 31;
  const int r0 = prow * 8, c0 = pcol * 8;

  // ---- stage 10x10 halo tile via async copy (coalesced channel-last reads);
  //      `ok` is uniform per pixel, so EXEC stays full for the async issues.
  for (int p = 0; p < 100; p++) {
    const int rr = r0 + (p / 10) - 1, cc = c0 + (p % 10) - 1;
    const bool ok = (rr >= 0) & (rr < HH) & (cc >= 0) & (cc < WW);
    if (ok) {
      async_g2lds_b32(y2 + ((size_t)(b * HH + rr) * WW + cc) * CMID + t,
                      ts + p * 257 + t);
    } else {
      ts[p * 257 + t] = 0.f;  // SAME zero padding
    }
  }
  for (int i = t; i < K2_WD; i += 256) wd[i] = w_dw[i];
  wait_async0();
  __syncthreads();

  // ---- dwconv + GLU: thread -> (pixel = t&63, j-block = t>>6), 32 j each ----
  {
    const int pix = t & 63, jb = t >> 6;
    const int pr = pix >> 3, pc = pix & 7;  // halo coords are +1
    for (int jj = 0; jj < 32; jj++) {
      const int j = jb * 32 + jj;  // gate channel 0..127; pair j+128
      float d1 = 0.f, d2 = 0.f;
#pragma unroll
      for (int di = 0; di < 3; di++)
#pragma unroll
        for (int dj = 0; dj < 3; dj++) {
          const int pn = (pr + di) * 10 + (pc + dj);
          d1 += wd[j * 9 + di * 3 + dj] * ts[pn * 257 + j];
          d2 += wd[(j + 128) * 9 + di * 3 + dj] * ts[pn * 257 + j + 128];
        }
      const float g = 0.5f * d1 * (1.f + erff(d1 * 0.70710678118654752f)) * d2;
      gs[j * 65 + pix] = g;
    }
  }
  __syncthreads();

  // ---- project_out GEMM: out[64][64pix] = w_out[64][128] * g[128][64] ----
  const int wave = t >> 5, lane = t & 31;
  const int lhi = lane >> 4, llo = lane & 15;
  const int mt = wave & 3, nb = (wave >> 2) * 2;  // wave: 1 M-tile x 2 N-tiles
  v8f acc2[2];
#pragma unroll
  for (int ni = 0; ni < 2; ni++)
#pragma unroll
    for (int r = 0; r < 8; r++) acc2[ni][r] = 0.f;

  for (int kk = 0; kk < 32; kk++) {
    const int M0 = mt * 16 + llo;
    const float* wp = w_out + M0 * CHID + kk * 4 + 2 * lhi;
    v2f a;
    a.x = wp[0];
    a.y = wp[1];
#pragma unroll
    for (int ni = 0; ni < 2; ni++) {
      const int N0 = (nb + ni) * 16 + llo;
      v2f bb;
      bb.x = gs[(kk * 4 + lhi) * 65 + N0];
      bb.y = gs[(kk * 4 + 2 + lhi) * 65 + N0];
      acc2[ni] = wmma4(a, bb, acc2[ni]);
    }
  }
#pragma unroll
  for (int ni = 0; ni < 2; ni++)
#pragma unroll
    for (int r = 0; r < 8; r++) {
      const int M = mt * 16 + r + 8 * lhi;
      const int p = (nb + ni) * 16 + llo;
      out[((size_t)(b * COUT + M) * HH + (r0 + (p >> 3))) * WW + (c0 + (p & 7))] =
          acc2[ni][r];
    }
}

// ============================ Launcher =====================================
extern "C" void kernel_launch(void* const* d_in, const int* in_sizes, int n_in,
                              void* d_out, int out_size, void* d_ws, size_t ws_size,
                              hipStream_t stream) {
  const float* x = (const float*)d_in[0];
  const float* w_in = (const float*)d_in[1];
  const float* w_dw = (const float*)d_in[2];
  const float* w_out = (const float*)d_in[3];
  const float* fft_w = (const float*)d_in[4];
  const float* alpha = (const float*)d_in[5];
  const float* beta = (const float*)d_in[6];
  float* out = (float*)d_out;
  float* y2 = (float*)d_ws;  // [B][H][W][256] channel-last, 256 MiB

  const dim3 grid(4 * 32 * 32);  // (batch, 8x8 patch)
  const dim3 block(256);
  k1_projin_spectral<<<grid, block, K1_LDS_F * sizeof(float), stream>>>(
      x, w_in, fft_w, alpha, beta, y2);
  k2_dw_glu_projout<<<grid, block, K2_LDS_F * sizeof(float), stream>>>(
      y2, w_dw, w_out, out);
}